// MinkConvBNRelu_82669530513900
// MI455X (gfx1250) — compile-verified
//
#include <hip/hip_runtime.h>

// MinkowskiEngine-style sparse conv + BN + ReLU for MI455X (gfx1250).
//
// Strategy (see analysis): memory-bound gather-GEMM -> keep fp32, use the
// CDNA5-only V_WMMA_F32_16X16X4_F32 for the matrix core. One wave32 owns a
// 16-row x 32-cout output tile; W is staged transposed in LDS with a padded
// stride (434 dwords: stride mod 64 = 50 -> 16 distinct even banks, so the
// per-lane ds_load_b64 B-fragment fetches are bank-conflict free and 8-byte
// aligned). BN stats are fused via LDS ds_add_f32 + global f32 atomics.

typedef __attribute__((ext_vector_type(2))) float v2f;
typedef __attribute__((ext_vector_type(8))) float v8f;

#define KOFF 27
#define CIN  16
#define COUT 32
#define KDIM (KOFF * CIN)      // 432 = GEMM reduction length
#define WT_STRIDE 434          // padded LDS row stride in dwords
#define WAVES_PER_BLOCK 8
#define BLOCK 256
#define BN_EPS 1e-5f

// ---------------------------------------------------------------------------
// Kernel 0: zero the 64-float sum/sumsq accumulator in workspace (d_ws is
// poisoned by the harness and never re-zeroed between graph replays).
// ---------------------------------------------------------------------------
__global__ void mink_zero_sums(float* __restrict__ sums) {
  sums[threadIdx.x] = 0.0f;  // launched with 64 threads
}

// ---------------------------------------------------------------------------
// Kernel 1: gather + WMMA GEMM + partial BN statistics.
//   conv[n, o] = sum_{k,c} mask[n,k] * feats[idx[n,k], c] * W[k, c, o]
// One wave per 16-row tile; 8 waves (256 threads) per block.
// ---------------------------------------------------------------------------
__global__ __launch_bounds__(BLOCK) void mink_conv_wmma(
    const float* __restrict__ feats,          // [N, 16]
    const float* __restrict__ Wk,             // [27, 16, 32] row-major
    const int* __restrict__ nidx,             // [N, 27]
    const unsigned char* __restrict__ nmask,  // [N, 27] (bool, 1 byte)
    float* __restrict__ conv,                 // [N, 32] output (pre-BN)
    float* __restrict__ sums,                 // [64]: sum[0:32], sumsq[32:64]
    int n)
{
  __shared__ float wt[COUT * WT_STRIDE];  // W transposed: wt[o][q], q = k*16+c
  __shared__ float red[64];               // block-level sum / sumsq reduction

  const int tid = threadIdx.x;

  // Stage W transposed into LDS (coalesced global reads).
  for (int i = tid; i < KDIM * COUT; i += BLOCK) {
    const int q = i >> 5;        // reduction index 0..431
    const int o = i & (COUT - 1);
    wt[o * WT_STRIDE + q] = Wk[i];
  }
  if (tid < 64) red[tid] = 0.0f;
  __syncthreads();

  const int wave = tid >> 5;
  const int lane = tid & 31;
  const int ln   = lane & 15;   // A: row-in-tile;  B/C: column (cout low 4b)
  const int hi   = lane >> 4;   // selects K-slot pair {0,1} vs {2,3}

  const int tile = blockIdx.x * WAVES_PER_BLOCK + wave;
  const bool active = (tile * 16) < n;   // wave-uniform: EXEC all-ones inside

  v8f acc0 = {};  // cout 0..15
  v8f acc1 = {};  // cout 16..31

  if (active) {
    const int row = tile * 16 + ln;      // this lane's A-matrix row
    const bool rok = row < n;

    for (int k = 0; k < KOFF; ++k) {
      int iv = 0;
      unsigned char mv = 0;
      if (rok) {
        iv = nidx[(size_t)row * KOFF + k];
        mv = nmask[(size_t)row * KOFF + k];
      }
      const float* src = feats + (size_t)iv * CIN;
      const bool use = rok && (mv != 0);

#pragma unroll
      for (int j = 0; j < 4; ++j) {
        // Global reduction index handled by this lane's two K-slots.
        const int q = (k << 4) + (j << 2) + (hi << 1);
        // A fragment: lanes 0-15 hold K-slots {0,1}; lanes 16-31 {2,3};
        // element = gathered[row, q] and [row, q+1] (8B aligned vector load).
        v2f a;
        if (use) {
          a = *(const v2f*)(src + (j << 2) + (hi << 1));
        } else {
          a.x = 0.0f; a.y = 0.0f;
        }
        // B fragments from LDS: B[Kslot, o] mirrored layout -> wt[o][q], wt[o][q+1].
        const v2f b0 = *(const v2f*)&wt[ln * WT_STRIDE + q];
        const v2f b1 = *(const v2f*)&wt[(ln + 16) * WT_STRIDE + q];

        acc0 = __builtin_amdgcn_wmma_f32_16x16x4_f32(
            false, a, false, b0, (short)0, acc0, false, false);
        acc1 = __builtin_amdgcn_wmma_f32_16x16x4_f32(
            false, a, false, b1, (short)0, acc1, false, false);
      }
    }

    // C/D layout: VGPR r, lanes 0-15 -> M=r, lanes 16-31 -> M=r+8; N = ln.
    float s0 = 0.f, s1 = 0.f, q0 = 0.f, q1 = 0.f;
#pragma unroll
    for (int r = 0; r < 8; ++r) {
      const int orow = tile * 16 + (hi << 3) + r;
      if (orow < n) {
        const float v0 = acc0[r];
        const float v1 = acc1[r];
        conv[(size_t)orow * COUT + ln]      = v0;
        conv[(size_t)orow * COUT + 16 + ln] = v1;
        s0 += v0; q0 += v0 * v0;
        s1 += v1; q1 += v1 * v1;
      }
    }
    // Per-channel partials into LDS (ds_add_f32), channels: ln and ln+16.
    atomicAdd(&red[ln],      s0);
    atomicAdd(&red[16 + ln], s1);
    atomicAdd(&red[32 + ln], q0);
    atomicAdd(&red[48 + ln], q1);
  }

  __syncthreads();
  if (tid < 64) {
    atomicAdd(&sums[tid], red[tid]);  // [0:32] sum, [32:64] sumsq
  }
}

// ---------------------------------------------------------------------------
// Kernel 2: reduce sums -> per-channel affine (scale, shift). 32 threads.
// ---------------------------------------------------------------------------
__global__ void mink_bn_finalize(const float* __restrict__ sums,
                                 const float* __restrict__ gamma,
                                 const float* __restrict__ beta,
                                 float* __restrict__ scale_shift,  // [64]
                                 float inv_n) {
  const int c = threadIdx.x;  // 0..31
  const float mean = sums[c] * inv_n;
  const float var  = sums[32 + c] * inv_n - mean * mean;
  const float inv  = rsqrtf(var + BN_EPS);
  const float sc   = gamma[c] * inv;
  scale_shift[c]      = sc;
  scale_shift[32 + c] = beta[c] - mean * sc;
}

// ---------------------------------------------------------------------------
// Kernel 3: in-place normalize + ReLU (float4; conv is L2-resident).
// ---------------------------------------------------------------------------
__global__ __launch_bounds__(BLOCK) void mink_bn_apply(
    float* __restrict__ out, const float* __restrict__ scale_shift, int total4) {
  const int i = blockIdx.x * BLOCK + threadIdx.x;
  if (i >= total4) return;
  float4 v = ((const float4*)out)[i];
  const int c = (i << 2) & (COUT - 1);  // channel of .x; .y/.z/.w follow
  v.x = fmaxf(0.f, v.x * scale_shift[c]     + scale_shift[32 + c]);
  v.y = fmaxf(0.f, v.y * scale_shift[c + 1] + scale_shift[32 + c + 1]);
  v.z = fmaxf(0.f, v.z * scale_shift[c + 2] + scale_shift[32 + c + 2]);
  v.w = fmaxf(0.f, v.w * scale_shift[c + 3] + scale_shift[32 + c + 3]);
  ((float4*)out)[i] = v;
}

// ---------------------------------------------------------------------------
// Launch: inputs in setup_inputs() order:
//   d_in[0]=feats f32[N,16], d_in[1]=W f32[27,16,32], d_in[2]=gamma f32[32],
//   d_in[3]=beta f32[32], d_in[4]=neighbor_idx i32[N,27],
//   d_in[5]=neighbor_mask bool(u8)[N,27].  d_out = f32[N,32].
// ---------------------------------------------------------------------------
extern "C" void kernel_launch(void* const* d_in, const int* in_sizes, int n_in,
                              void* d_out, int out_size, void* d_ws, size_t ws_size,
                              hipStream_t stream) {
  const float* feats = (const float*)d_in[0];
  const float* Wk    = (const float*)d_in[1];
  const float* gamma = (const float*)d_in[2];
  const float* beta  = (const float*)d_in[3];
  const int* nidx    = (const int*)d_in[4];
  const unsigned char* nmask = (const unsigned char*)d_in[5];

  float* conv = (float*)d_out;
  float* sums = (float*)d_ws;            // [0:64) sum/sumsq
  float* scsh = (float*)d_ws + 64;       // [64:128) scale/shift

  const int n = in_sizes[0] / CIN;       // 500000
  const int tiles = (n + 15) / 16;
  const int blocks = (tiles + WAVES_PER_BLOCK - 1) / WAVES_PER_BLOCK;

  mink_zero_sums<<<1, 64, 0, stream>>>(sums);
  mink_conv_wmma<<<blocks, BLOCK, 0, stream>>>(feats, Wk, nidx, nmask,
                                               conv, sums, n);
  mink_bn_finalize<<<1, 32, 0, stream>>>(sums, gamma, beta, scsh, 1.0f / (float)n);
  const int total4 = (n * COUT) / 4;
  mink_bn_apply<<<(total4 + BLOCK - 1) / BLOCK, BLOCK, 0, stream>>>(conv, scsh, total4);
}